// LossFactory_57604101373978
// MI455X (gfx1250) — compile-verified
//
#include <hip/hip_runtime.h>
#include <math.h>

// ArcFace-style loss: -mean_i [ num_i - logsumexp_j(logits_ij) ]
//   logits_ij = S*wf[i,j] (j != label_i),  num_i = S*(wf[i,label_i] - M)
// Memory-bound single pass: 524 MB @ 23.3 TB/s => ~22.5 us floor.
// CDNA5 path: global_load_async_to_lds_b128 double-buffered staging + s_wait_asynccnt.

#define SCALE_S 30.0f
#define MARGIN_M 0.4f
#define BLOCK 256
#define CHUNKV 1024                 // float4 vectors per chunk (16 KB / buffer)
#define ISSUE (CHUNKV / BLOCK)      // async b128 ops per thread per chunk
static_assert(ISSUE == 4, "s_wait_asynccnt literals below assume 4 ops/wave/chunk");

__device__ __forceinline__ void lse_merge(float& m, float& s, float om, float os) {
    float nm = fmaxf(m, om);
    s = s * __expf(m - nm) + os * __expf(om - nm);
    m = nm;
}

// Issue exactly ISSUE async b128 global->LDS copies per thread (uniform per-wave
// ASYNCcnt). Out-of-range lanes are clamped to the last vector: duplicate LDS
// writes of identical data are harmless, and the wait literal stays exact.
__device__ __forceinline__ void issue_chunk(const float* __restrict__ rowbase,
                                            float* __restrict__ ldsbuf,
                                            int vbase, int cvecs) {
    const int tid = threadIdx.x;
#pragma unroll
    for (int i = 0; i < ISSUE; ++i) {
        int v  = tid + i * BLOCK;
        int vc = (v < cvecs) ? v : (cvecs - 1);
        const float* g = rowbase + (size_t)(vbase + vc) * 4u;
        unsigned lds   = (unsigned)(size_t)(ldsbuf + (size_t)vc * 4u); // AS3 offset = low 32 bits
        asm volatile("global_load_async_to_lds_b128 %0, %1, off"
                     :: "v"(lds), "v"(g) : "memory");
    }
}

__global__ void __launch_bounds__(BLOCK)
arcface_row_lse(const float* __restrict__ wf, const long long* __restrict__ labels,
                float* __restrict__ row_loss, int C) {
    __shared__ float buf[2][CHUNKV * 4];     // 2 x 16 KB staging buffers
    __shared__ float wm[BLOCK / 32], ws_[BLOCK / 32];

    const int row = blockIdx.x;
    const int tid = threadIdx.x;
    const float* rowbase = wf + (size_t)row * (size_t)C;

    const int nvec   = C >> 2;                         // C % 4 == 0 (32000)
    const int nchunk = (nvec + CHUNKV - 1) / CHUNKV;

    float target = 0.0f;
    if (tid == 0) target = rowbase[(int)labels[row]];  // raw cosine of label column

    float m = -INFINITY, s = 0.0f;

    issue_chunk(rowbase, &buf[0][0], 0, min(CHUNKV, nvec));

    for (int c = 0; c < nchunk; ++c) {
        const int cvecs = min(CHUNKV, nvec - c * CHUNKV);
        if (c + 1 < nchunk) {
            issue_chunk(rowbase, &buf[(c + 1) & 1][0], (c + 1) * CHUNKV,
                        min(CHUNKV, nvec - (c + 1) * CHUNKV));
            asm volatile("s_wait_asynccnt 0x4" ::: "memory");  // chunk c complete, c+1 in flight
        } else {
            asm volatile("s_wait_asynccnt 0x0" ::: "memory");
        }
        __syncthreads();   // all waves' async portions landed -> whole chunk readable

        const float4* lb = (const float4*)&buf[c & 1][0];
        for (int v = tid; v < cvecs; v += BLOCK) {
            float4 f = lb[v];
            float e4[4] = {f.x, f.y, f.z, f.w};
#pragma unroll
            for (int k = 0; k < 4; ++k) {
                float x  = SCALE_S * e4[k];
                float nm = fmaxf(m, x);
                float e  = __expf(fminf(m, x) - nm);   // one exp per element
                s = (x > m) ? fmaf(s, e, 1.0f) : (s + e);
                m = nm;
            }
        }
        __syncthreads();   // done reading buf[c&1] before it is re-issued at c+2
    }

    // wave32 reduction of (max, sum) pairs
#pragma unroll
    for (int off = 16; off > 0; off >>= 1) {
        float om = __shfl_xor(m, off);
        float os = __shfl_xor(s, off);
        lse_merge(m, s, om, os);
    }
    if ((tid & 31) == 0) { wm[tid >> 5] = m; ws_[tid >> 5] = s; }
    __syncthreads();

    if (tid == 0) {
        float M = wm[0], Ssum = ws_[0];
        for (int w = 1; w < BLOCK / 32; ++w) lse_merge(M, Ssum, wm[w], ws_[w]);
        // swap the label column's contribution: raw S*t -> S*(t - M_margin)
        float num     = SCALE_S * (target - MARGIN_M);
        float sum_adj = Ssum - __expf(SCALE_S * target - M) + __expf(num - M);
        float log_den = M + __logf(sum_adj);
        row_loss[row] = num - log_den;
    }
}

__global__ void __launch_bounds__(BLOCK)
mean_neg(const float* __restrict__ row_loss, float* __restrict__ out, int B) {
    __shared__ float sm[BLOCK / 32];
    float acc = 0.0f;
    for (int i = threadIdx.x; i < B; i += BLOCK) acc += row_loss[i];
#pragma unroll
    for (int off = 16; off > 0; off >>= 1) acc += __shfl_xor(acc, off);
    if ((threadIdx.x & 31) == 0) sm[threadIdx.x >> 5] = acc;
    __syncthreads();
    if (threadIdx.x == 0) {
        float t = 0.0f;
        for (int w = 0; w < BLOCK / 32; ++w) t += sm[w];
        out[0] = -t / (float)B;
    }
}

extern "C" void kernel_launch(void* const* d_in, const int* in_sizes, int n_in,
                              void* d_out, int out_size, void* d_ws, size_t ws_size,
                              hipStream_t stream) {
    const float*     wf     = (const float*)d_in[0];
    const long long* labels = (const long long*)d_in[1];   // int64 labels
    float* out      = (float*)d_out;
    float* row_loss = (float*)d_ws;                        // B floats of scratch

    const int B = in_sizes[1];              // 4096
    const int C = in_sizes[0] / B;          // 32000

    arcface_row_lse<<<B, BLOCK, 0, stream>>>(wf, labels, row_loss, C);
    mean_neg<<<1, BLOCK, 0, stream>>>(row_loss, out, B);
}